// Polarisation_68865505624311
// MI455X (gfx1250) — compile-verified
//
#include <hip/hip_runtime.h>
#include <stdint.h>
#include <stddef.h>

// Thole-damped dipole field:  out = mu/pol_au + segment_sum(T_ij mu_j)
// Memory-bound: 24B/edge streamed (154MB), gathers/atomics L2-resident.
// CDNA5 path: async global->LDS double-buffered pipeline (ASYNCcnt).

#define TILE 1024   // edges per tile (per block)
#define TPB  256    // threads per block; 4 edges per thread per tile

static constexpr float kBohr = 0.52917721067f;
static constexpr float kA    = 0.39f;

#if defined(__has_builtin)
#if __has_builtin(__builtin_amdgcn_global_load_async_to_lds_b128)
#define ASYNC_EN 1
#endif
#endif
#ifndef ASYNC_EN
#define ASYNC_EN 0
#endif

#if ASYNC_EN
typedef int v4i_t __attribute__((ext_vector_type(4)));
typedef v4i_t __attribute__((address_space(1))) g_v4i;
typedef v4i_t __attribute__((address_space(3))) l_v4i;
__device__ __forceinline__ void async_b128(const void* g, void* l) {
  __builtin_amdgcn_global_load_async_to_lds_b128((g_v4i*)g, (l_v4i*)l, 0, 0);
}
#endif

__device__ __forceinline__ void edge_contrib(
    int s, int d, float dist_a, float vx, float vy, float vz,
    const float* __restrict__ pol, const float* __restrict__ mu,
    float* __restrict__ out)
{
  const float invB  = 1.0f / kBohr;
  const float invB3 = invB * invB * invB;

  float r  = dist_a * invB;            // rij (au)
  float ex = vx * invB;                // vec_ij (au)
  float ey = vy * invB;
  float ez = vz * invB;

  float ps = pol[s] * invB3;           // pol in au
  float pd = pol[d] * invB3;
  float alpha = ps * pd;

  float r2 = r * r;
  float r3 = r2 * r;
  // uij^3 = r^3 * alpha^(-1/2)  ->  single v_rsq_f32, no pow()
  float au3  = kA * r3 * __builtin_amdgcn_rsqf(alpha);
  float et   = __expf(-au3);           // v_exp_f32
  float lam3 = 1.0f - et;
  float lam5 = 1.0f - (1.0f + au3) * et;

  float inv_r  = __builtin_amdgcn_rcpf(r);
  float inv_r2 = inv_r * inv_r;
  float inv_r3 = inv_r2 * inv_r;
  float inv_r5 = inv_r3 * inv_r2;

  float mx = mu[3 * (size_t)d + 0];    // L2-resident gather
  float my = mu[3 * (size_t)d + 1];
  float mz = mu[3 * (size_t)d + 2];

  float dot = ex * mx + ey * my + ez * mz;
  float c3  = lam3 * inv_r3;
  float c5  = 3.0f * lam5 * inv_r5 * dot;

  float fx = c3 * mx - c5 * ex;
  float fy = c3 * my - c5 * ey;
  float fz = c3 * mz - c5 * ez;

  // non-returning fp32 atomics -> L2 atomic units
  (void)__hip_atomic_fetch_add(&out[3 * (size_t)s + 0], fx, __ATOMIC_RELAXED, __HIP_MEMORY_SCOPE_AGENT);
  (void)__hip_atomic_fetch_add(&out[3 * (size_t)s + 1], fy, __ATOMIC_RELAXED, __HIP_MEMORY_SCOPE_AGENT);
  (void)__hip_atomic_fetch_add(&out[3 * (size_t)s + 2], fz, __ATOMIC_RELAXED, __HIP_MEMORY_SCOPE_AGENT);
}

// Diagonal term: out[i] = mu[i] * BOHR^3 / pol[i]  (full overwrite of d_out)
__global__ void __launch_bounds__(TPB) Polarisation_diag_kernel(
    const float* __restrict__ pol, const float* __restrict__ mu,
    float* __restrict__ out, int nNodes)
{
  int i = blockIdx.x * blockDim.x + threadIdx.x;
  if (i < nNodes) {
    const float B3 = kBohr * kBohr * kBohr;
    float inv_p = B3 / pol[i];
    out[3 * (size_t)i + 0] = mu[3 * (size_t)i + 0] * inv_p;
    out[3 * (size_t)i + 1] = mu[3 * (size_t)i + 1] * inv_p;
    out[3 * (size_t)i + 2] = mu[3 * (size_t)i + 2] * inv_p;
  }
}

__global__ void __launch_bounds__(TPB) Polarisation_edge_kernel(
    const int* __restrict__ esrc, const int* __restrict__ edst,
    const float* __restrict__ dist, const float* __restrict__ vec,
    const float* __restrict__ pol, const float* __restrict__ mu,
    float* __restrict__ out, int nEdges)
{
  const int tid = (int)threadIdx.x;
  const int numTiles = nEdges / TILE;

#if ASYNC_EN
  __shared__ __align__(16) int   s_src [2][TILE];
  __shared__ __align__(16) int   s_dst [2][TILE];
  __shared__ __align__(16) float s_dist[2][TILE];
  __shared__ __align__(16) float s_vec [2][TILE * 3];

  const int e4 = tid * 4;  // 16B-aligned lane slice of the tile

  auto stage = [&](int t, int b) {
    const int base = t * TILE;
    async_b128(&esrc[base + e4], &s_src [b][e4]);
    async_b128(&edst[base + e4], &s_dst [b][e4]);
    async_b128(&dist[base + e4], &s_dist[b][e4]);
    const float* vb = vec + (size_t)(base + e4) * 3;   // 48B per thread, 16B aligned
    async_b128(vb + 0, &s_vec[b][e4 * 3 + 0]);
    async_b128(vb + 4, &s_vec[b][e4 * 3 + 4]);
    async_b128(vb + 8, &s_vec[b][e4 * 3 + 8]);
  };

  int t   = (int)blockIdx.x;
  int buf = 0;
  if (t < numTiles) stage(t, 0);
  for (; t < numTiles; t += (int)gridDim.x) {
    const int tn = t + (int)gridDim.x;
    if (tn < numTiles) {
      stage(tn, buf ^ 1);                               // 6 async ops in flight for next tile
      asm volatile("s_wait_asynccnt 6" ::: "memory");   // current tile's 6 ops done (in-order)
    } else {
      asm volatile("s_wait_asynccnt 0" ::: "memory");
    }
#pragma unroll
    for (int k = 0; k < 4; ++k) {
      const int j = e4 + k;
      edge_contrib(s_src[buf][j], s_dst[buf][j], s_dist[buf][j],
                   s_vec[buf][3 * j + 0], s_vec[buf][3 * j + 1], s_vec[buf][3 * j + 2],
                   pol, mu, out);
    }
    buf ^= 1;
  }
#else
  // Fallback: direct streaming loads (still coalesced; cache handles it)
  for (int t = (int)blockIdx.x; t < numTiles; t += (int)gridDim.x) {
    const int base = t * TILE + tid * 4;
#pragma unroll
    for (int k = 0; k < 4; ++k) {
      const int j = base + k;
      edge_contrib(esrc[j], edst[j], dist[j],
                   vec[3 * (size_t)j + 0], vec[3 * (size_t)j + 1], vec[3 * (size_t)j + 2],
                   pol, mu, out);
    }
  }
#endif

  // Remainder edges (nEdges not multiple of TILE)
  for (int i = numTiles * TILE + (int)(blockIdx.x * blockDim.x) + tid;
       i < nEdges; i += (int)(gridDim.x * blockDim.x)) {
    edge_contrib(esrc[i], edst[i], dist[i],
                 vec[3 * (size_t)i + 0], vec[3 * (size_t)i + 1], vec[3 * (size_t)i + 2],
                 pol, mu, out);
  }
}

extern "C" void kernel_launch(void* const* d_in, const int* in_sizes, int n_in,
                              void* d_out, int out_size, void* d_ws, size_t ws_size,
                              hipStream_t stream) {
  (void)n_in; (void)out_size; (void)d_ws; (void)ws_size;
  // inputs: 0 species(i32) 1 edge_src(i32) 2 edge_dst(i32) 3 distances(f32)
  //         4 vec(f32) 5 polarisability(f32) 6 mu(f32)
  const int*   esrc = (const int*)  d_in[1];
  const int*   edst = (const int*)  d_in[2];
  const float* dist = (const float*)d_in[3];
  const float* vec  = (const float*)d_in[4];
  const float* pol  = (const float*)d_in[5];
  const float* mu   = (const float*)d_in[6];
  float*       out  = (float*)d_out;

  const int nEdges = in_sizes[1];
  const int nNodes = in_sizes[5];

  // 1) diagonal term initializes out (stream order guarantees it precedes atomics)
  Polarisation_diag_kernel<<<(nNodes + TPB - 1) / TPB, TPB, 0, stream>>>(pol, mu, out, nNodes);

  // 2) edge scatter-add
  int numTiles = nEdges / TILE;
  int blocks = numTiles < 1 ? 1 : (numTiles > 2048 ? 2048 : numTiles);
  Polarisation_edge_kernel<<<blocks, TPB, 0, stream>>>(esrc, edst, dist, vec, pol, mu, out, nEdges);
}